// gradient_attention_17712445129406
// MI455X (gfx1250) — compile-verified
//
#include <hip/hip_runtime.h>
#include <math.h>

// Problem shape (fixed by the reference): grad is [N, D] fp32, row-major.
#define N 8192
#define D 4096
#define TEMP_INV (1.0f / 100.0f)
#define CH 32            // row chunks for two-stage column reductions
#define RPC (N / CH)     // rows per chunk = 256

typedef __attribute__((ext_vector_type(2))) float v2f;
typedef __attribute__((ext_vector_type(8))) float v8f;
typedef __attribute__((ext_vector_type(4))) int   v4i;

#define AS1 __attribute__((address_space(1)))
#define AS3 __attribute__((address_space(3)))

#if __has_builtin(__builtin_amdgcn_global_load_async_to_lds_b128) && \
    __has_builtin(__builtin_amdgcn_s_wait_asynccnt)
#define USE_ASYNC 1
#else
#define USE_ASYNC 0
#endif

// ---------------------------------------------------------------------------
// Pass 1 / Pass 4: (weighted) partial column sums.
//   partial[chunk][col] = sum_{r in chunk} (useW ? w[r] : 1) * G[r][col]
// Fully coalesced: consecutive threads read consecutive columns.
// ---------------------------------------------------------------------------
__global__ __launch_bounds__(256)
void col_partial_kernel(const float* __restrict__ grad,
                        const float* __restrict__ w,
                        float* __restrict__ partial, int useW) {
    const int col   = blockIdx.x * 256 + threadIdx.x;
    const int chunk = blockIdx.y;
    const int row0  = chunk * RPC;
    const float* g  = grad + (size_t)row0 * D + col;
    float acc = 0.0f;
#pragma unroll 4
    for (int r = 0; r < RPC; ++r) {
        float v = g[(size_t)r * D];
        acc += useW ? (w[row0 + r] * v) : v;
    }
    partial[(size_t)chunk * D + col] = acc;
}

// partial[CH][D] -> dst[D], fixed summation order (deterministic)
__global__ __launch_bounds__(256)
void reduce_chunks_kernel(const float* __restrict__ partial,
                          float* __restrict__ dst) {
    const int col = blockIdx.x * 256 + threadIdx.x;
    float s = 0.0f;
#pragma unroll
    for (int c = 0; c < CH; ++c) s += partial[(size_t)c * D + col];
    dst[col] = s;
}

// ---------------------------------------------------------------------------
// Pass 2: scores[i] = G_i . colsum via V_WMMA_F32_16X16X4_F32.
// One block = 16 rows. Double-buffered LDS tiles staged with
// GLOBAL_LOAD_ASYNC_TO_LDS_B128 (prefetch tile t+1 while tile t feeds WMMA).
// 8 waves each own a 32-wide K-slice of a 256-wide tile (8 WMMAs each).
//
// ISA fragment layouts (f32, wave32):
//   A 16x4 : lane L<16 -> M=L, VGPR0=K0, VGPR1=K1 ; lane L+16 -> K2,K3
//   B 4x16 : mirror (VGPR j, lanes<16 -> K=j ; lanes>=16 -> K=2+j)
//   D 16x16: VGPR v, lanes<16 -> M=v, N=lane ; lanes>=16 -> M=v+8
// ---------------------------------------------------------------------------
#define KT  256           // K tile width staged in LDS
#define KTP (KT + 4)      // padded stride: rows land on distinct banks
#define NT  (D / KT)      // 16 K-tiles
#define TILE_F (16 * KTP) // floats per buffer

__device__ __forceinline__
void stage_tile(const float* __restrict__ grad, int rowBase, int kb,
                float* lbuf, int tid) {
    // 16 rows x KT floats = (16*KT/4) float4 transfers, 256 threads
#if USE_ASYNC
#pragma unroll
    for (int i = 0; i < (16 * KT / 4) / 256; ++i) {
        int idx = tid + i * 256;            // float4 index
        int r   = idx / (KT / 4);
        int k4  = idx % (KT / 4);
        const float* g = grad + (size_t)(rowBase + r) * D + kb + k4 * 4;
        float*       l = lbuf + r * KTP + k4 * 4;
        __builtin_amdgcn_global_load_async_to_lds_b128(
            (AS1 v4i*)g, (AS3 v4i*)l, /*offset=*/0, /*cpol=*/0);
    }
#else
    float4 v[(16 * KT / 4) / 256];
#pragma unroll
    for (int i = 0; i < (16 * KT / 4) / 256; ++i) {
        int idx = tid + i * 256;
        int r   = idx / (KT / 4);
        int k4  = idx % (KT / 4);
        v[i] = *(reinterpret_cast<const float4*>(
            grad + (size_t)(rowBase + r) * D + kb) + k4);
    }
#pragma unroll
    for (int i = 0; i < (16 * KT / 4) / 256; ++i) {
        int idx = tid + i * 256;
        int r   = idx / (KT / 4);
        int k4  = idx % (KT / 4);
        *reinterpret_cast<float4*>(&lbuf[r * KTP + k4 * 4]) = v[i];
    }
#endif
}

__device__ __forceinline__ void wait_stage() {
#if USE_ASYNC
    __builtin_amdgcn_s_wait_asynccnt(0);
#endif
}

__global__ __launch_bounds__(256)
void scores_wmma_kernel(const float* __restrict__ grad,
                        const float* __restrict__ colsum,
                        float* __restrict__ scores) {
    __shared__ float tile[2 * TILE_F];   // ~33 KB double buffer
    __shared__ float wpart[8][16];

    const int rowBase = blockIdx.x * 16;
    const int tid  = threadIdx.x;
    const int wave = tid >> 5;
    const int lane = tid & 31;
    const int h    = lane >> 4;          // half-wave select (K pair)
    const int l16  = lane & 15;
    const int kw   = wave * (KT / 8);    // 32 K-values per wave per tile

    v8f acc = {};

    // Prologue: stage tile 0
    stage_tile(grad, rowBase, 0, tile, tid);

    for (int t = 0; t < NT; ++t) {
        wait_stage();
        __syncthreads();                 // tile[t&1] ready; other buf free

        if (t + 1 < NT)                  // prefetch next tile into other buf
            stage_tile(grad, rowBase, (t + 1) * KT, tile + ((t + 1) & 1) * TILE_F, tid);

        const float* tb = tile + (t & 1) * TILE_F;
        const int kb = t * KT;

        // Hoist all fragment loads so they clause ahead of the WMMA chain.
        v2f afrag[KT / 32], bfrag[KT / 32];
#pragma unroll
        for (int j = 0; j < KT / 32; ++j) {
            const int kl = kw + 4 * j + 2 * h;   // K offset within tile
            const int kg = kb + kl;              // global K
            afrag[j].x = tb[l16 * KTP + kl];
            afrag[j].y = tb[l16 * KTP + kl + 1];
            bfrag[j].x = colsum[kg];
            bfrag[j].y = colsum[kg + 1];
        }
#pragma unroll
        for (int j = 0; j < KT / 32; ++j) {
            acc = __builtin_amdgcn_wmma_f32_16x16x4_f32(
                /*neg_a=*/false, afrag[j], /*neg_b=*/false, bfrag[j],
                /*c_mod=*/(short)0, acc, /*reuse_a=*/false, /*reuse_b=*/false);
        }
        __syncthreads();                 // all reads of tile[t&1] done
    }

    // Column 0 of each half-wave carries the 16 row results.
    if (l16 == 0) {
#pragma unroll
        for (int v = 0; v < 8; ++v) wpart[wave][h * 8 + v] = acc[v];
    }
    __syncthreads();
    // Deterministic fixed-order cross-wave combine.
    if (tid < 16) {
        float s = 0.0f;
#pragma unroll
        for (int wv = 0; wv < 8; ++wv) s += wpart[wv][tid];
        scores[rowBase + tid] = s;
    }
}

// ---------------------------------------------------------------------------
// Pass 3: w = softmax(scores / 100). Single block (n = 8192 is tiny).
// ---------------------------------------------------------------------------
__global__ __launch_bounds__(1024)
void softmax_kernel(const float* __restrict__ scores,
                    float* __restrict__ w) {
    __shared__ float red[1024];
    const int tid = threadIdx.x;

    float m = -INFINITY;
    for (int i = tid; i < N; i += 1024) m = fmaxf(m, scores[i]);
    red[tid] = m;
    __syncthreads();
    for (int s = 512; s > 0; s >>= 1) {
        if (tid < s) red[tid] = fmaxf(red[tid], red[tid + s]);
        __syncthreads();
    }
    m = red[0];
    __syncthreads();

    float sum = 0.0f;
    for (int i = tid; i < N; i += 1024)
        sum += expf((scores[i] - m) * TEMP_INV);
    red[tid] = sum;
    __syncthreads();
    for (int s = 512; s > 0; s >>= 1) {
        if (tid < s) red[tid] += red[tid + s];
        __syncthreads();
    }
    const float inv = 1.0f / red[0];

    for (int i = tid; i < N; i += 1024)
        w[i] = expf((scores[i] - m) * TEMP_INV) * inv;
}

// ---------------------------------------------------------------------------
// Launch: colsum (2-stage) -> scores (WMMA + async staging) -> softmax -> pool
// Workspace floats: partial[CH*D] | colsum[D] | scores[N] | w[N]  (~600 KB)
// ---------------------------------------------------------------------------
extern "C" void kernel_launch(void* const* d_in, const int* in_sizes, int n_in,
                              void* d_out, int out_size, void* d_ws, size_t ws_size,
                              hipStream_t stream) {
    (void)in_sizes; (void)n_in; (void)out_size; (void)ws_size;
    const float* grad = (const float*)d_in[0];
    float* out     = (float*)d_out;
    float* ws      = (float*)d_ws;
    float* partial = ws;                       // CH*D
    float* colsum  = partial + (size_t)CH * D; // D
    float* scores  = colsum + D;               // N
    float* w       = scores + N;               // N

    // Pass 1: colsum
    col_partial_kernel<<<dim3(D / 256, CH), 256, 0, stream>>>(grad, nullptr, partial, 0);
    reduce_chunks_kernel<<<D / 256, 256, 0, stream>>>(partial, colsum);
    // Pass 2: scores via f32 WMMA with async double-buffered staging
    scores_wmma_kernel<<<N / 16, 256, 0, stream>>>(grad, colsum, scores);
    // Pass 3: softmax weights
    softmax_kernel<<<1, 1024, 0, stream>>>(scores, w);
    // Pass 4: weighted pooling
    col_partial_kernel<<<dim3(D / 256, CH), 256, 0, stream>>>(grad, w, partial, 1);
    reduce_chunks_kernel<<<D / 256, 256, 0, stream>>>(partial, out);
}